// SupportVectorRegression_20779051778152
// MI455X (gfx1250) — compile-verified
//
#include <hip/hip_runtime.h>

typedef __attribute__((ext_vector_type(2))) float v2f;
typedef __attribute__((ext_vector_type(8))) float v8f;

#define NSAMP 4096
#define DFEAT 64
#define TILE 16
#define KSTEPS (DFEAT / 4)      // 16 wmma f32 steps of K=4
#define NTILES (NSAMP / TILE)   // 256

// exp(-d2 / (2*sigma^2)), sigma = 5
__device__ __constant__ float kCoef = -1.0f / (2.0f * 5.0f * 5.0f);

// ---------------- kernel 1: per-row squared norms ----------------
__global__ void svr_sq_kernel(const float* __restrict__ f, float* __restrict__ sq) {
    int i = blockIdx.x * blockDim.x + threadIdx.x;
    if (i >= NSAMP) return;
    const float4* row = (const float4*)(f + (size_t)i * DFEAT);
    float s = 0.f;
#pragma unroll
    for (int q = 0; q < DFEAT / 4; ++q) {
        float4 v = row[q];
        s = fmaf(v.x, v.x, s);
        s = fmaf(v.y, v.y, s);
        s = fmaf(v.z, v.z, s);
        s = fmaf(v.w, v.w, s);
    }
    sq[i] = s;
}

// ---------------- kernel 2: fused WMMA gram tile + exp + alpha-dot ----------------
// One wave per (i-tile, j-chunk). FP32 WMMA 16x16x4, K=64 -> 16 chained wmma.
// Never materializes the 4096x4096 kernel matrix.
__global__ void __launch_bounds__(32)
svr_tile_kernel(const float* __restrict__ f, const float* __restrict__ sq,
                const float* __restrict__ alpha, float* __restrict__ partial,
                int tilesPerChunk) {
    const int lane = threadIdx.x;        // 0..31, wave32
    const int i0   = blockIdx.x * TILE;  // row tile base
    const int chunk = blockIdx.y;        // j-range split
    const int row  = lane & 15;          // M for A frag / N for B frag / col for C
    const int koff = (lane >> 4) * 2;    // K half selection within a 4-wide step

    // ---- A fragments (16x64 strip) held in registers for the whole j loop ----
    v2f a[KSTEPS];
    {
        const float* arow = f + (size_t)(i0 + row) * DFEAT + koff;
#pragma unroll
        for (int s = 0; s < KSTEPS; ++s)
            a[s] = *(const v2f*)(arow + 4 * s);
    }

    // per-accumulator-register row norms: VGPR r holds row (r + 8*(lane>>4))
    float sqi[8];
#pragma unroll
    for (int r = 0; r < 8; ++r)
        sqi[r] = sq[i0 + r + ((lane >> 4) << 3)];

    v8f acc = {};   // per-row partial sums of exp(...) * alpha[j]

    const int jt0 = chunk * tilesPerChunk;
    for (int jt = jt0; jt < jt0 + tilesPerChunk; ++jt) {
        const int j0 = jt * TILE;

        // B fragments: identical load pattern (F * F^T symmetry of layouts)
        v2f b[KSTEPS];
        const float* brow = f + (size_t)(j0 + row) * DFEAT + koff;
#pragma unroll
        for (int s = 0; s < KSTEPS; ++s)
            b[s] = *(const v2f*)(brow + 4 * s);

        const float sqj = sq[j0 + row];      // column (lane&15) of this tile
        const float alj = alpha[j0 + row];

        // 16x16 gram tile: 16 chained v_wmma_f32_16x16x4_f32
        v8f c = {};
#pragma unroll
        for (int s = 0; s < KSTEPS; ++s)
            c = __builtin_amdgcn_wmma_f32_16x16x4_f32(
                    /*neg_a=*/false, a[s], /*neg_b=*/false, b[s],
                    /*c_mod=*/(short)0, c, /*reuse_a=*/false, /*reuse_b=*/false);

        // fused epilogue: d2 -> exp -> * alpha, accumulate per row
#pragma unroll
        for (int r = 0; r < 8; ++r) {
            float d2 = fmaxf(fmaf(-2.0f, c[r], sqi[r] + sqj), 0.0f);
            acc[r] = fmaf(__expf(d2 * kCoef), alj, acc[r]);
        }
    }

    // reduce across the 16 lanes of each half (halves are independent rows)
#pragma unroll
    for (int r = 0; r < 8; ++r) {
        float v = acc[r];
        v += __shfl_xor(v, 8, 32);
        v += __shfl_xor(v, 4, 32);
        v += __shfl_xor(v, 2, 32);
        v += __shfl_xor(v, 1, 32);
        if ((lane & 15) == 0) {
            int gi = i0 + r + ((lane >> 4) << 3);
            partial[(size_t)chunk * NSAMP + gi] = v;
        }
    }
}

// ---------------- kernel 3: deterministic chunk reduction + bias ----------------
__global__ void svr_reduce_kernel(const float* __restrict__ partial,
                                  const float* __restrict__ bias,
                                  float* __restrict__ out, int nchunks) {
    int i = blockIdx.x * blockDim.x + threadIdx.x;
    if (i >= NSAMP) return;
    float s = bias[0];
    for (int c = 0; c < nchunks; ++c)
        s += partial[(size_t)c * NSAMP + i];
    out[i] = s;
}

extern "C" void kernel_launch(void* const* d_in, const int* in_sizes, int n_in,
                              void* d_out, int out_size, void* d_ws, size_t ws_size,
                              hipStream_t stream) {
    (void)in_sizes; (void)n_in; (void)out_size;
    const float* f     = (const float*)d_in[0];   // [4096, 64]
    const float* alpha = (const float*)d_in[1];   // [4096, 1]
    const float* bias  = (const float*)d_in[2];   // [1, 1]
    float* out = (float*)d_out;                   // [4096, 1]

    // workspace layout: sq[N] | partial[JC * N]
    int jc = 8;
    size_t need8 = (size_t)NSAMP * sizeof(float) * (1 + 8);
    if (ws_size < need8) jc = 1;   // deterministic fallback for small workspaces
    int tilesPerChunk = NTILES / jc;

    float* sq      = (float*)d_ws;
    float* partial = sq + NSAMP;

    svr_sq_kernel<<<(NSAMP + 255) / 256, 256, 0, stream>>>(f, sq);

    dim3 grid(NTILES, jc);
    svr_tile_kernel<<<grid, 32, 0, stream>>>(f, sq, alpha, partial, tilesPerChunk);

    svr_reduce_kernel<<<(NSAMP + 255) / 256, 256, 0, stream>>>(partial, bias, out, jc);
}